// RegionProposalNetwork_39109972197419
// MI455X (gfx1250) — compile-verified
//
#include <hip/hip_runtime.h>
#include <stdint.h>

#define NBINS     8192      // 13-bit radix histogram
#define CAND_CAP  8192      // candidate capacity per image (power of 2 for bitonic)
#define PRE_K     6000      // PRE_NMS
#define POST_K    1000      // POST_NMS
#define NMS_TH    0.7f
#define MIN_SZ    1.0f
#define BBOX_CLIP 4.135166556742356f   // log(1000/16)

typedef unsigned int v4u __attribute__((ext_vector_type(4)));
typedef int          v8i __attribute__((ext_vector_type(8)));
typedef int          v4i __attribute__((ext_vector_type(4)));

// Map float to monotonically ordered u32 (total order, descending sortable).
__device__ __forceinline__ uint32_t rpn_fkey(float f) {
  uint32_t u = __float_as_uint(f);
  return (u & 0x80000000u) ? ~u : (u | 0x80000000u);
}

// ---------------- workspace zeroing ----------------
__global__ void rpn_init_ws(uint32_t* __restrict__ ws32, int nWords) {
  int i = blockIdx.x * blockDim.x + threadIdx.x;
  int stride = gridDim.x * blockDim.x;
  for (; i < nWords; i += stride) ws32[i] = 0u;
}

// ------- pass 1: per-image 8192-bin histogram (grid.y = image, no divides) --
__global__ void rpn_hist(const float* __restrict__ obj, int N,
                         uint32_t* __restrict__ hist) {
  int b = blockIdx.y;
  const float* o = obj + (size_t)b * N;
  uint32_t* h = hist + (size_t)b * NBINS;
  int stride = gridDim.x * blockDim.x;
  for (int i = blockIdx.x * blockDim.x + threadIdx.x; i < N; i += stride)
    atomicAdd(&h[rpn_fkey(o[i]) >> 19], 1u);   // top 13 bits
}

// ---------------- pass 2: find threshold bin per image ----------------------
__global__ void rpn_threshold(const uint32_t* __restrict__ hist,
                              uint32_t* __restrict__ thrBin,
                              uint32_t* __restrict__ strictCnt) {
  __shared__ uint32_t h[NBINS];
  int b = blockIdx.x;
  for (int t = threadIdx.x; t < NBINS; t += blockDim.x)
    h[t] = hist[(size_t)b * NBINS + t];
  __syncthreads();
  if (threadIdx.x == 0) {
    uint32_t cum = 0, strict = 0;
    int T = 0; bool found = false;
    for (int bin = NBINS - 1; bin >= 0; --bin) {
      uint32_t c = h[bin];
      if (cum + c >= (uint32_t)PRE_K) { T = bin; strict = cum; found = true; break; }
      cum += c;
    }
    if (!found) { T = 0; strict = cum - h[0]; }   // degenerate: take everything
    thrBin[b] = (uint32_t)T;
    strictCnt[b] = strict;                         // keys in bins > T (< PRE_K)
  }
}

// ---------------- pass 3: compact candidates >= threshold bin ---------------
// Strict-above entries fill from the front (provably < PRE_K of them);
// threshold-bin ties fill from the back. Middle gap stays zero (sorts last).
__global__ void rpn_compact(const float* __restrict__ obj, int N,
                            const uint32_t* __restrict__ thrBin,
                            const uint32_t* __restrict__ strictCnt,
                            uint32_t* __restrict__ cntA,
                            uint32_t* __restrict__ cntB,
                            uint64_t* __restrict__ candKeys) {
  int b = blockIdx.y;
  const float* o = obj + (size_t)b * N;
  uint64_t* dst = candKeys + (size_t)b * CAND_CAP;
  uint32_t T = thrBin[b];
  uint32_t strict = strictCnt[b];
  int stride = gridDim.x * blockDim.x;
  for (int j = blockIdx.x * blockDim.x + threadIdx.x; j < N; j += stride) {
    uint32_t key = rpn_fkey(o[j]);
    uint32_t bin = key >> 19;
    if (bin < T) continue;
    // composite: descending sort => score desc, then index asc (top_k ties)
    uint64_t comp = ((uint64_t)key << 32) | (uint64_t)(~(uint32_t)j);
    if (bin > T) {
      uint32_t pos = atomicAdd(&cntA[b], 1u);
      if (pos < CAND_CAP) dst[pos] = comp;
    } else {
      uint32_t cap = CAND_CAP - strict;
      uint32_t pos = atomicAdd(&cntB[b], 1u);
      if (pos < cap) dst[CAND_CAP - 1 - pos] = comp;
    }
  }
}

// ---------------- pass 4: per-image sort + decode + NMS + emit --------------
// One 1024-thread (32-wave) workgroup per image; ~209 KB of the 320 KB WGP LDS.
__global__ __launch_bounds__(1024, 1)
void rpn_per_image(const float* __restrict__ anchors,
                   const float* __restrict__ deltas,
                   const uint64_t* __restrict__ candKeys,
                   int N,
                   const int* __restrict__ imgHp,
                   const int* __restrict__ imgWp,
                   float* __restrict__ out) {
  extern __shared__ char smem[];
  uint64_t* keys  = (uint64_t*)smem;                                       // 64 KB
  float*    bx    = (float*)(smem + CAND_CAP * 8);                         // 96000 B
  float*    areas = (float*)(smem + CAND_CAP * 8 + PRE_K * 16);            // 24000 B
  uint32_t* flags = (uint32_t*)(smem + CAND_CAP * 8 + PRE_K * 16 + PRE_K * 4);
  uint32_t* scanb = (uint32_t*)(smem + CAND_CAP * 8 + PRE_K * 16 + PRE_K * 8);

  const int tid  = threadIdx.x;
  const int nthr = blockDim.x;
  const int b    = blockIdx.x;
  const float imgH = (float)imgHp[0];
  const float imgW = (float)imgWp[0];

  const uint64_t* gk = candKeys + (size_t)b * CAND_CAP;
  const uint32_t ldsKeys = (uint32_t)(uintptr_t)(void*)keys;

  // --- stage candidate keys (64 KB contiguous) into LDS --------------------
#if __has_builtin(__builtin_amdgcn_tensor_load_to_lds) && __has_builtin(__builtin_amdgcn_s_wait_tensorcnt)
  // CDNA5 Tensor Data Mover: one wave issues a 1-row 8192x8B tile DMA.
  if (tid < 32) {
    uint64_t ga = (uint64_t)(uintptr_t)gk;
    v4u g0;
    g0.x = 1u;                                        // count=1, user mode
    g0.y = ldsKeys;                                   // lds_addr
    g0.z = (uint32_t)ga;                              // global_addr[95:64]
    g0.w = ((uint32_t)(ga >> 32) & 0x01FFFFFFu)       // global_addr[120:96]
         | (2u << 30);                                // type = 2 ("image")
    v8i g1;
    g1[0] = (int)(3u << 16);                          // data_size = 8 bytes
    g1[1] = (int)((uint32_t)CAND_CAP << 16);          // tensor_dim0[15:0] @ [63:48]
    g1[2] = (int)(1u << 16);                          // dim0 hi=0; tensor_dim1=1
    g1[3] = (int)((uint32_t)CAND_CAP << 16);          // dim1 hi=0; tile_dim0=8192
    g1[4] = 1;                                        // tile_dim1=1, tile_dim2=0
    g1[5] = CAND_CAP;                                 // tensor_dim0_stride lo32
    g1[6] = 0;                                        // stride hi; dim1_stride lo
    g1[7] = 0;
    v4i gz4 = {0, 0, 0, 0};
    v8i gz8 = {0, 0, 0, 0, 0, 0, 0, 0};
    // 6-arg variant (clang-23 / therock-10.0): groups 2/3 unused -> zeros
    __builtin_amdgcn_tensor_load_to_lds(g0, g1, gz4, gz4, gz8, 0);
    __builtin_amdgcn_s_wait_tensorcnt(0);
  }
#else
  // Fallback: per-lane CDNA5 async global->LDS copies.
  for (int t = tid; t < CAND_CAP; t += nthr) {
    uint32_t ld = ldsKeys + (uint32_t)t * 8u;
    uint64_t ga = (uint64_t)(uintptr_t)(gk + t);
    asm volatile("global_load_async_to_lds_b64 %0, %1, off"
                 :: "v"(ld), "v"(ga) : "memory");
  }
  asm volatile("s_wait_asynccnt 0" ::: "memory");
#endif
  __syncthreads();

  // --- bitonic sort 8192 u64 composites, descending -----------------------
  for (int k = 2; k <= CAND_CAP; k <<= 1) {
    for (int s = k >> 1; s > 0; s >>= 1) {
      __syncthreads();
      for (int t = tid; t < CAND_CAP / 2; t += nthr) {
        int i = ((t & ~(s - 1)) << 1) | (t & (s - 1));
        int p = i + s;
        uint64_t a = keys[i], c = keys[p];
        bool desc = ((i & k) == 0);
        if (desc ? (a < c) : (a > c)) { keys[i] = c; keys[p] = a; }
      }
    }
  }
  __syncthreads();

  // --- decode / clip / validate top PRE_K boxes into LDS -------------------
  const float4* anc = (const float4*)anchors + (size_t)b * N;
  const float4* del = (const float4*)deltas  + (size_t)b * N;
  for (int t = tid; t < PRE_K; t += nthr) {
    uint64_t key = keys[t];
    float4 box = make_float4(0.f, 0.f, 0.f, 0.f);
    float area = 0.f;
    uint32_t fl = 0u;
    if (key != 0ull) {
      uint32_t idx = ~(uint32_t)(key & 0xFFFFFFFFull);
      float4 a = anc[idx];
      float4 d = del[idx];
      float wa = a.z - a.x, ha = a.w - a.y;
      float cxa = a.x + 0.5f * wa, cya = a.y + 0.5f * ha;
      float dw = fminf(d.z, BBOX_CLIP), dh = fminf(d.w, BBOX_CLIP);
      float cx = d.x * wa + cxa, cy = d.y * ha + cya;
      float w = expf(dw) * wa, h = expf(dh) * ha;
      float x1 = fminf(fmaxf(cx - 0.5f * w, 0.f), imgW);
      float y1 = fminf(fmaxf(cy - 0.5f * h, 0.f), imgH);
      float x2 = fminf(fmaxf(cx + 0.5f * w, 0.f), imgW);
      float y2 = fminf(fmaxf(cy + 0.5f * h, 0.f), imgH);
      box = make_float4(x1, y1, x2, y2);
      area = (x2 - x1) * (y2 - y1);
      fl = ((x2 - x1) >= MIN_SZ && (y2 - y1) >= MIN_SZ) ? 1u : 0u;
    }
    ((float4*)bx)[t] = box;
    areas[t] = area;
    flags[t] = fl;
  }
  __syncthreads();

  // --- greedy NMS, boxes + flags resident in LDS ---------------------------
  for (int i = 0; i < PRE_K - 1; ++i) {
    if (flags[i]) {                       // uniform LDS broadcast read
      float4 bi = ((const float4*)bx)[i];
      float ai = areas[i];
      for (int j = i + 1 + tid; j < PRE_K; j += nthr) {
        if (flags[j]) {
          float4 bj = ((const float4*)bx)[j];
          float xx1 = fmaxf(bi.x, bj.x);
          float yy1 = fmaxf(bi.y, bj.y);
          float xx2 = fminf(bi.z, bj.z);
          float yy2 = fminf(bi.w, bj.w);
          float inter = fmaxf(xx2 - xx1, 0.f) * fmaxf(yy2 - yy1, 0.f);
          float iou = inter / (ai + areas[j] - inter + 1e-9f);
          if (iou > NMS_TH) flags[j] = 0u;
        }
      }
    }
    __syncthreads();
  }

  // --- rank kept boxes (block prefix scan) and pack output in LDS ----------
  // keys region (64 KB) is dead after decode: reuse first 16 KB as out buffer
  float* outLds = (float*)keys;
  int carry = 0;
  for (int c = 0; c < (PRE_K + 1023) / 1024; ++c) {
    int pos = c * 1024 + tid;
    uint32_t v = (pos < PRE_K) ? flags[pos] : 0u;
    scanb[tid] = v;
    __syncthreads();
    for (int off = 1; off < 1024; off <<= 1) {
      uint32_t add = (tid >= off) ? scanb[tid - off] : 0u;
      __syncthreads();
      scanb[tid] += add;
      __syncthreads();
    }
    int rank = carry + (int)scanb[tid] - 1;
    if (pos < PRE_K && v && rank < POST_K)
      ((float4*)outLds)[rank] = ((const float4*)bx)[pos];
    carry += (int)scanb[1023];
    __syncthreads();
  }

  // zero-fill tail in LDS (disjoint from scattered ranks)
  int kept = carry < POST_K ? carry : POST_K;
  for (int t = kept * 4 + tid; t < POST_K * 4; t += nthr) outLds[t] = 0.f;
  __syncthreads();

  // --- drain packed output LDS -> HBM with CDNA5 async stores (16 KB) ------
  {
    float* outB = out + (size_t)b * POST_K * 4;
    uint32_t ldsOut = (uint32_t)(uintptr_t)(void*)outLds;
    for (int t = tid; t < POST_K; t += nthr) {
      uint32_t ld = ldsOut + (uint32_t)t * 16u;
      uint64_t ga = (uint64_t)(uintptr_t)(outB + t * 4);
      asm volatile("global_store_async_from_lds_b128 %0, %1, off"
                   :: "v"(ga), "v"(ld) : "memory");
    }
    asm volatile("s_wait_asynccnt 0" ::: "memory");
  }
}

extern "C" void kernel_launch(void* const* d_in, const int* in_sizes, int n_in,
                              void* d_out, int out_size, void* d_ws, size_t ws_size,
                              hipStream_t stream) {
  (void)n_in; (void)ws_size;
  const float* anchors = (const float*)d_in[0];
  const float* obj     = (const float*)d_in[1];
  const float* deltas  = (const float*)d_in[2];
  const int*   imgh    = (const int*)d_in[3];
  const int*   imgw    = (const int*)d_in[4];
  float* out = (float*)d_out;

  int B = out_size / (POST_K * 4);          // 8
  int N = in_sizes[1] / B;                  // 500000

  // workspace (u32 words): hist | thrBin | strictCnt | cntA | cntB | candKeys(u64)
  uint32_t* ws32      = (uint32_t*)d_ws;
  uint32_t* hist      = ws32;
  uint32_t* thrBin    = ws32 + (size_t)B * NBINS;
  uint32_t* strictCnt = thrBin + B;
  uint32_t* cntA      = strictCnt + B;
  uint32_t* cntB      = cntA + B;
  size_t off = (size_t)B * NBINS + 4 * (size_t)B;
  off = (off + 1) & ~(size_t)1;             // 8-byte align for u64 keys
  uint64_t* candKeys = (uint64_t*)(ws32 + off);

  int zeroWords = (int)(off + (size_t)B * CAND_CAP * 2);
  rpn_init_ws<<<1024, 256, 0, stream>>>(ws32, zeroWords);

  int bpi = (N + 255) / 256;                // blocks per image
  if (bpi > 2048) bpi = 2048;
  dim3 grid2(bpi, B);
  rpn_hist<<<grid2, 256, 0, stream>>>(obj, N, hist);
  rpn_threshold<<<B, 1024, 0, stream>>>(hist, thrBin, strictCnt);
  rpn_compact<<<grid2, 256, 0, stream>>>(obj, N, thrBin, strictCnt,
                                         cntA, cntB, candKeys);

  size_t lds = (size_t)CAND_CAP * 8      // sort keys / packed output
             + (size_t)PRE_K * 16        // boxes
             + (size_t)PRE_K * 4         // areas
             + (size_t)PRE_K * 4         // flags
             + 1024 * 4;                 // scan buffer  => 213632 B (< 320 KB)
  rpn_per_image<<<B, 1024, lds, stream>>>(anchors, deltas, candKeys, N,
                                          imgh, imgw, out);
}